// Net_68702296866895
// MI455X (gfx1250) — compile-verified
//
#include <hip/hip_runtime.h>
#include <math.h>

#define B_GRAPHS 1024
#define NPG 258
#define NPG_PAD 272           // pad to 17 * 16 for WMMA tiles
#define F_IN 64
#define DEG 8
#define EPG (NPG * DEG)       // 2064 edges per graph
#define CH 32
#define HID 128
#define KD1 (NPG * CH)        // 8256
#define N_NODES (B_GRAPHS * NPG)
#define KSPLIT 4
#define KSEG (KD1 / KSPLIT)   // 2064
#define KCHUNK 516            // KSEG / 4, multiple of 4
#define Y1SZ (B_GRAPHS * HID) // one partial buffer

typedef __attribute__((ext_vector_type(2))) float v2f;
typedef __attribute__((ext_vector_type(8))) float v8f;

__device__ __forceinline__ v8f wmma_f32x4(v2f a, v2f b, v8f c) {
    // D = A(16x4) * B(4x16) + C, full fp32 -> v_wmma_f32_16x16x4_f32
    return __builtin_amdgcn_wmma_f32_16x16x4_f32(
        /*neg_a=*/false, a, /*neg_b=*/false, b,
        /*c_mod=*/(short)0, c, /*reuse_a=*/false, /*reuse_b=*/false);
}

// ---------------------------------------------------------------------------
// Kernel 1: per-graph ChebConv (K=2) fused: LDS-staged x, LDS-atomic scatter
// for ax = A_hat x, then WMMA GEMM  h = elu(x@W0 + ax@W1 + b)
// One workgroup (256 threads = 8 waves) per graph.
// Dynamic LDS: xs[272*64] | axs[272*64] | w0s[64*32] | w1s[64*32]  (~152 KB)
// ---------------------------------------------------------------------------
__global__ void cheb_conv_kernel(const float* __restrict__ x,
                                 const int* __restrict__ edge_src,
                                 const int* __restrict__ edge_dst,
                                 const float* __restrict__ edge_w,
                                 const float* __restrict__ W0,
                                 const float* __restrict__ W1,
                                 const float* __restrict__ b_conv,
                                 float* __restrict__ h) {
    extern __shared__ float smem[];
    float* xs  = smem;                        // NPG_PAD * 64
    float* axs = smem + NPG_PAD * F_IN;       // NPG_PAD * 64
    float* w0s = axs + NPG_PAD * F_IN;        // 64 * 32
    float* w1s = w0s + F_IN * CH;             // 64 * 32

    const int g   = blockIdx.x;
    const int tid = threadIdx.x;

    const int total_f = NPG_PAD * F_IN * 2 + F_IN * CH * 2;
    for (int i = tid; i < total_f; i += 256) smem[i] = 0.0f;
    __syncthreads();

    // stage x for this graph (rows are contiguous: g*258*64)
    const float* xg = x + (size_t)g * NPG * F_IN;
    for (int i = tid; i < NPG * F_IN; i += 256) xs[i] = xg[i];
    for (int i = tid; i < F_IN * CH; i += 256) { w0s[i] = W0[i]; w1s[i] = W1[i]; }
    __syncthreads();

    // edge scatter-add into LDS: 16 lanes per edge, 4 feats per lane (float4)
    const int ebase  = g * EPG;
    const int goff   = g * NPG;
    const int grp    = tid >> 4;      // 0..15
    const int lane16 = tid & 15;
    const int f0     = lane16 * 4;
    for (int e = grp; e < EPG; e += 16) {
        const int   s = edge_src[ebase + e] - goff;
        const int   d = edge_dst[ebase + e] - goff;
        const float w = edge_w[ebase + e];
        const float4 xv = *(const float4*)&xs[s * F_IN + f0];
        atomicAdd(&axs[d * F_IN + f0 + 0], w * xv.x);
        atomicAdd(&axs[d * F_IN + f0 + 1], w * xv.y);
        atomicAdd(&axs[d * F_IN + f0 + 2], w * xv.z);
        atomicAdd(&axs[d * F_IN + f0 + 3], w * xv.w);
    }
    __syncthreads();

    // WMMA GEMM: 17 M-tiles x 2 N-tiles, K = 64 in steps of 4
    const int wave = tid >> 5;
    const int lane = tid & 31;
    const int l15  = lane & 15;
    const int khalf = (lane < 16) ? 0 : 2;

    for (int t = wave; t < 17 * 2; t += 8) {
        const int m0 = (t % 17) * 16;
        const int n0 = (t / 17) * 16;
        const int arow = m0 + l15;
        const int col  = n0 + l15;
        v8f c = {};
        #pragma unroll
        for (int k0 = 0; k0 < F_IN; k0 += 4) {
            v2f a0, a1, b0, b1;
            a0.x = xs[arow * F_IN + k0 + khalf];
            a0.y = xs[arow * F_IN + k0 + khalf + 1];
            a1.x = axs[arow * F_IN + k0 + khalf];
            a1.y = axs[arow * F_IN + k0 + khalf + 1];
            b0.x = w0s[(k0 + khalf) * CH + col];
            b0.y = w0s[(k0 + khalf + 1) * CH + col];
            b1.x = w1s[(k0 + khalf) * CH + col];
            b1.y = w1s[(k0 + khalf + 1) * CH + col];
            c = wmma_f32x4(a0, b0, c);   // x @ W0
            c = wmma_f32x4(a1, b1, c);   // ax @ W1
        }
        const float bias = b_conv[col];
        const int rbase = m0 + ((lane >= 16) ? 8 : 0);
        #pragma unroll
        for (int i = 0; i < 8; ++i) {
            const int row = rbase + i;
            float v = c[i] + bias;
            v = (v > 0.0f) ? v : (__expf(v) - 1.0f);   // ELU
            if (row < NPG)
                h[((size_t)goff + row) * CH + col] = v;
        }
    }
}

// ---------------------------------------------------------------------------
// Kernel 2: partial GEMM for y1 = h[1024,8256] @ Wd1[8256,128]
// grid (64 M-tiles, 4 K-splits), 256 threads = 8 waves (one 16-col N tile each).
// A chunks (16 x 516, 33 KB) staged in LDS cooperatively; waves read A via ds,
// B streams from L2. Each K-split writes its own partial buffer (deterministic;
// no float atomics). Bias/relu deferred to the tail kernel.
// ---------------------------------------------------------------------------
__global__ void dense1_kernel(const float* __restrict__ h,
                              const float* __restrict__ Wd1,
                              float* __restrict__ y1p) {
    __shared__ float as_[16 * KCHUNK];        // 33 KB

    const int tid  = threadIdx.x;
    const int wave = tid >> 5;
    const int lane = tid & 31;
    const int l15  = lane & 15;
    const int khalf = (lane < 16) ? 0 : 2;

    const int m0 = blockIdx.x * 16;
    const int kb = blockIdx.y * KSEG;         // this block's K range
    const int n0 = wave * 16;
    const int col = n0 + l15;

    v8f c = {};
    for (int chunk = 0; chunk < KSEG / KCHUNK; ++chunk) {
        const int base = kb + chunk * KCHUNK;
        // cooperative float4 stage of A chunk: 16 rows x 516 = 2064 float4
        for (int i4 = tid; i4 < 16 * (KCHUNK / 4); i4 += 256) {
            const int row = i4 / (KCHUNK / 4);
            const int c4  = (i4 % (KCHUNK / 4)) * 4;
            *(float4*)&as_[row * KCHUNK + c4] =
                *(const float4*)&h[(size_t)(m0 + row) * KD1 + base + c4];
        }
        __syncthreads();
        #pragma unroll 8
        for (int k0 = 0; k0 < KCHUNK; k0 += 4) {
            v2f a, b;
            a.x = as_[l15 * KCHUNK + k0 + khalf];
            a.y = as_[l15 * KCHUNK + k0 + khalf + 1];
            b.x = Wd1[(size_t)(base + k0 + khalf) * HID + col];
            b.y = Wd1[(size_t)(base + k0 + khalf + 1) * HID + col];
            c = wmma_f32x4(a, b, c);
        }
        __syncthreads();
    }
    float* outp = y1p + (size_t)blockIdx.y * Y1SZ;
    const int rbase = m0 + ((lane >= 16) ? 8 : 0);
    #pragma unroll
    for (int i = 0; i < 8; ++i)
        outp[(size_t)(rbase + i) * HID + col] = c[i];
}

// ---------------------------------------------------------------------------
// Kernel 3: tail MLP. Sums the 4 K-split partials + bd1 (+relu) into LDS,
// then 128 -> 64 -> 32 -> 1 (+sigmoid) entirely out of LDS.
// 16 blocks x 64 threads (one row per thread). 56 KB static LDS.
// ---------------------------------------------------------------------------
__global__ void tail_kernel(const float* __restrict__ y1p,
                            const float* __restrict__ bd1,
                            const float* __restrict__ Wd2, const float* __restrict__ bd2,
                            const float* __restrict__ Wd3, const float* __restrict__ bd3,
                            const float* __restrict__ Wd4, const float* __restrict__ bd4,
                            float* __restrict__ out) {
    __shared__ float t1[64 * 128];
    __shared__ float t2[64 * 64];
    __shared__ float t3[64 * 32];
    const int tid = threadIdx.x;
    const int r   = blockIdx.x * 64 + tid;

    for (int k = 0; k < HID; ++k) {
        const size_t idx = (size_t)r * HID + k;
        float s = y1p[idx] + y1p[idx + Y1SZ] + y1p[idx + 2 * (size_t)Y1SZ]
                + y1p[idx + 3 * (size_t)Y1SZ] + bd1[k];
        t1[tid * 128 + k] = fmaxf(s, 0.0f);   // relu(layer-1)
    }
    for (int j = 0; j < 64; ++j) {
        float acc = bd2[j];
        #pragma unroll 8
        for (int k = 0; k < HID; ++k) acc += t1[tid * 128 + k] * Wd2[k * 64 + j];
        t2[tid * 64 + j] = fmaxf(acc, 0.0f);
    }
    for (int j = 0; j < 32; ++j) {
        float acc = bd3[j];
        #pragma unroll 8
        for (int k = 0; k < 64; ++k) acc += t2[tid * 64 + k] * Wd3[k * 32 + j];
        t3[tid * 32 + j] = fmaxf(acc, 0.0f);
    }
    float acc = bd4[0];
    #pragma unroll
    for (int k = 0; k < 32; ++k) acc += t3[tid * 32 + k] * Wd4[k];
    out[r] = 1.0f / (1.0f + __expf(-acc));
}

// ---------------------------------------------------------------------------
extern "C" void kernel_launch(void* const* d_in, const int* in_sizes, int n_in,
                              void* d_out, int out_size, void* d_ws, size_t ws_size,
                              hipStream_t stream) {
    const float* x    = (const float*)d_in[0];
    const int*   esrc = (const int*)  d_in[1];
    const int*   edst = (const int*)  d_in[2];
    const float* ew   = (const float*)d_in[3];
    const float* W0   = (const float*)d_in[4];
    const float* W1   = (const float*)d_in[5];
    const float* bc   = (const float*)d_in[6];
    const float* Wd1  = (const float*)d_in[7];
    const float* bd1  = (const float*)d_in[8];
    const float* Wd2  = (const float*)d_in[9];
    const float* bd2  = (const float*)d_in[10];
    const float* Wd3  = (const float*)d_in[11];
    const float* bd3  = (const float*)d_in[12];
    const float* Wd4  = (const float*)d_in[13];
    const float* bd4  = (const float*)d_in[14];
    float* out = (float*)d_out;

    float* h   = (float*)d_ws;                      // [N_NODES, 32]   33.8 MB
    float* y1p = h + (size_t)N_NODES * CH;          // [4, 1024, 128]   2.0 MB

    const size_t smem = (size_t)(NPG_PAD * F_IN * 2 + F_IN * CH * 2) * sizeof(float);
    cheb_conv_kernel<<<B_GRAPHS, 256, smem, stream>>>(x, esrc, edst, ew, W0, W1, bc, h);
    dense1_kernel<<<dim3(B_GRAPHS / 16, KSPLIT), 256, 0, stream>>>(h, Wd1, y1p);
    tail_kernel<<<B_GRAPHS / 64, 64, 0, stream>>>(y1p, bd1, Wd2, bd2, Wd3, bd3,
                                                  Wd4, bd4, out);
}